// DenseCapsule_13065290515054
// MI455X (gfx1250) — compile-verified
//
#include <hip/hip_runtime.h>
#include <math.h>

// ---------------------------------------------------------------------------
// CapsNet dynamic routing, fully fused for MI455X (gfx1250, wave32).
//   x      : [B=64, I=2048, Din=8]          fp32
//   weight : [O=64, I=2048, Dout=16, Din=8] fp32
//   out    : [B=64, O=64, Dout=16]          fp32
//
// Strategy: never materialize x_hat (512 MB). Recompute it per routing
// iteration with V_WMMA_F32_16X16X4_F32; weight (64 MB) stays L2-resident.
// Logits are linear in the accumulated outputs, so routing state is just
// acc_out [B,O,16] (256 KB) instead of b [B,O,I] (32 MB).
// Weight slices are streamed into double-buffered LDS with the gfx1250
// async-to-LDS DMA path (GLOBAL_LOAD_ASYNC_TO_LDS_B128 + s_wait_asynccnt),
// overlapping the L2 fetch of slice i+1 with WMMA+routing of slice i.
// ---------------------------------------------------------------------------

typedef float v2f __attribute__((ext_vector_type(2)));
typedef float v8f __attribute__((ext_vector_type(8)));

#define B_      64
#define I_      2048
#define DIN     8
#define O_      64
#define DOUT    16
#define BB      16                  // batches per workgroup (= WMMA M tile)
#define NCHUNK  32                  // i-chunks (grid.x)
#define ICHUNK  (I_ / NCHUNK)       // 64 i per workgroup
#define NTHREADS 1024               // 32 waves
#define WSLICE  (O_ * DOUT * DIN)   // 8192 floats = 32 KB per i
#define CAPS_EPS 1e-8f

// ---------------------------------------------------------------------------
__global__ __launch_bounds__(NTHREADS)
void caps_route_pass(const float* __restrict__ x,
                     const float* __restrict__ w,
                     const float* __restrict__ acc,
                     float* __restrict__ sp)
{
    __shared__ float wlds[2][WSLICE];        // 64 KB  double-buffered W slice
    __shared__ float xh[BB * O_ * DOUT];     // 64 KB  x_hat tile  [b][o][d]
    __shared__ float xi[BB * DIN];           // 512 B  x slice     [b][k]
    __shared__ float rmax[BB][2];            // per-row wave maxima
    __shared__ float rsum[BB][2];            // per-row wave exp-sums

    const int tid     = threadIdx.x;
    const int lane    = tid & 31;
    const int wv      = tid >> 5;            // wave id 0..31
    const int chunk   = blockIdx.x;
    const int bblock  = blockIdx.y;          // 0..3
    const int i0      = chunk * ICHUNK;
    const int bbase   = bblock * BB;

    const int halfsel = lane >> 4;           // 0 | 1  (wave32 half)
    const int lm      = lane & 15;           // lane % 16

    // routing identity: one thread per (b_local, o); row = 2 waves
    const int bl      = tid >> 6;            // 0..15
    const int oo      = tid & 63;            // 0..63
    const int rowhalf = wv & 1;              // which half of row bl

    // LDS byte offsets of the two weight buffers (flat-pointer low 32 bits
    // are the LDS byte address for shared allocations)
    const unsigned wbase0 = (unsigned)(uintptr_t)&wlds[0][0];
    const unsigned wbase1 = (unsigned)(uintptr_t)&wlds[1][0];

    // per-thread register state: acc_out dot vector + s accumulator
    float accv[DOUT];
    float sacc[DOUT];
    {
        const float* ap = acc + ((size_t)(bbase + bl) * O_ + oo) * DOUT;
        #pragma unroll
        for (int d = 0; d < DOUT; ++d) { accv[d] = ap[d]; sacc[d] = 0.0f; }
    }

    // async stage of W[:, i, :, :] into LDS buffer `bsel`: 2048 x b128,
    // two issues per thread -> ASYNCcnt += 2 per wave
    #define ASYNC_STAGE(ivar, baseoff)                                        \
    {                                                                         \
        _Pragma("unroll")                                                     \
        for (int r = 0; r < 2; ++r) {                                         \
            const int idx = tid + r * NTHREADS;     /* 0..2047 float4 */      \
            const int o4  = idx >> 5;               /* 32 float4 per o */     \
            const int rem = idx & 31;                                         \
            const float* srcp = w + (size_t)o4 * (I_ * DOUT * DIN)            \
                                  + (size_t)(ivar) * (DOUT * DIN) + rem * 4;  \
            const unsigned dstoff = (baseoff) + (unsigned)(idx * 16);         \
            asm volatile("global_load_async_to_lds_b128 %0, %1, off"          \
                         :: "v"(dstoff), "v"(srcp) : "memory");               \
        }                                                                     \
    }

    // prologue: kick off DMA for the first slice
    ASYNC_STAGE(i0, wbase0);

    for (int ii = 0; ii < ICHUNK; ++ii) {
        const int i = i0 + ii;
        const int buf = ii & 1;
        __syncthreads();  // prev iter done with xh/xi/rmax/rsum

        // issue DMA for the NEXT slice into the other buffer (its previous
        // readers finished two barriers ago), then wait for current buffer.
        if (ii + 1 < ICHUNK) {
            ASYNC_STAGE(i + 1, buf ? wbase0 : wbase1);
        }
        // stage x[bbase.., i, :] (tiny, regular path)
        if (tid < BB * DIN) {
            const int b = tid >> 3, k = tid & 7;
            xi[tid] = x[(size_t)(bbase + b) * (I_ * DIN) + (size_t)i * DIN + k];
        }
        if (ii + 1 < ICHUNK) {
            asm volatile("s_wait_asynccnt 0x2" ::: "memory"); // in-order: cur done
        } else {
            asm volatile("s_wait_asynccnt 0x0" ::: "memory");
        }
        __syncthreads();

        // --- WMMA: C(16b x 16d per o) = A(16b x 8k) * B(8k x 16d) ---------
        // A layout (32b 16x4): lane holds M = lane%16, K = 4s + 2*half + j
        const float* wbuf = &wlds[buf][0];
        v2f afrag[2];
        #pragma unroll
        for (int s = 0; s < 2; ++s) {
            #pragma unroll
            for (int j = 0; j < 2; ++j)
                afrag[s][j] = xi[lm * DIN + (4 * s + 2 * halfsel + j)];
        }

        #pragma unroll
        for (int t = 0; t < 2; ++t) {            // 2 output capsules / wave
            const int o = wv * 2 + t;
            v8f cacc = {0.f, 0.f, 0.f, 0.f, 0.f, 0.f, 0.f, 0.f};
            #pragma unroll
            for (int s = 0; s < 2; ++s) {
                v2f bfrag;                        // B: N = lane%16 (=d), K as A
                #pragma unroll
                for (int j = 0; j < 2; ++j)
                    bfrag[j] = wbuf[o * (DOUT * DIN) + lm * DIN
                                    + (4 * s + 2 * halfsel + j)];
                cacc = __builtin_amdgcn_wmma_f32_16x16x4_f32(
                           false, afrag[s], false, bfrag,
                           (short)0, cacc, false, false);
            }
            // C layout: row M = j + 8*half, col N = lane%16 (=d)
            #pragma unroll
            for (int j = 0; j < 8; ++j)
                xh[(j + 8 * halfsel) * (O_ * DOUT) + o * DOUT + lm] = cacc[j];
        }
        __syncthreads();

        // --- routing: logit, softmax over o (2-wave rows), accumulate s ---
        const float* xrow = xh + bl * (O_ * DOUT) + oo * DOUT;
        float logit = 0.f;
        #pragma unroll
        for (int d = 0; d < DOUT; ++d) logit = fmaf(xrow[d], accv[d], logit);

        // wave32 butterfly max over this wave's 32 o-values
        float m = logit;
        #pragma unroll
        for (int off = 16; off; off >>= 1)
            m = fmaxf(m, __shfl_xor(m, off, 32));
        if (lane == 0) rmax[bl][rowhalf] = m;
        __syncthreads();
        m = fmaxf(rmax[bl][0], rmax[bl][1]);

        const float e = __expf(logit - m);
        float es = e;
        #pragma unroll
        for (int off = 16; off; off >>= 1)
            es += __shfl_xor(es, off, 32);
        if (lane == 0) rsum[bl][rowhalf] = es;
        __syncthreads();
        const float c = e / (rsum[bl][0] + rsum[bl][1]);

        #pragma unroll
        for (int d = 0; d < DOUT; ++d) sacc[d] = fmaf(c, xrow[d], sacc[d]);
    }
    #undef ASYNC_STAGE

    // --- emit deterministic per-chunk partial s ---------------------------
    float* spp = sp + (((size_t)chunk * B_ + bbase + bl) * O_ + oo) * DOUT;
    #pragma unroll
    for (int d = 0; d < DOUT; ++d) spp[d] = sacc[d];
}

// ---------------------------------------------------------------------------
// Reduce partial s over chunks, squash; update acc_out or write final output.
// One thread per (b, o) pair: 4096 threads.
// ---------------------------------------------------------------------------
__global__ __launch_bounds__(256)
void caps_reduce_squash(const float* __restrict__ sp,
                        float* __restrict__ acc,
                        float* __restrict__ out,
                        int last)
{
    const int g = blockIdx.x * 256 + threadIdx.x;   // (b*64 + o)
    float s[DOUT];
    #pragma unroll
    for (int d = 0; d < DOUT; ++d) s[d] = 0.f;
    for (int ch = 0; ch < NCHUNK; ++ch) {
        const float* p = sp + ((size_t)ch * (B_ * O_) + g) * DOUT;
        #pragma unroll
        for (int d = 0; d < DOUT; ++d) s[d] += p[d];
    }
    float n2 = 0.f;
    #pragma unroll
    for (int d = 0; d < DOUT; ++d) n2 = fmaf(s[d], s[d], n2);
    const float norm  = sqrtf(n2);
    const float scale = n2 / (1.f + n2) / (norm + CAPS_EPS);

    if (last) {
        float* op = out + (size_t)g * DOUT;
        #pragma unroll
        for (int d = 0; d < DOUT; ++d) op[d] = scale * s[d];
    } else {
        float* ap = acc + (size_t)g * DOUT;
        #pragma unroll
        for (int d = 0; d < DOUT; ++d) ap[d] += scale * s[d];
    }
}

__global__ void caps_zero(float* __restrict__ p, int n)
{
    const int i = blockIdx.x * blockDim.x + threadIdx.x;
    if (i < n) p[i] = 0.f;
}

// ---------------------------------------------------------------------------
extern "C" void kernel_launch(void* const* d_in, const int* in_sizes, int n_in,
                              void* d_out, int out_size, void* d_ws, size_t ws_size,
                              hipStream_t stream)
{
    const float* x = (const float*)d_in[0];   // [64, 2048, 8]
    const float* w = (const float*)d_in[1];   // [64, 2048, 16, 8]
    float* outp = (float*)d_out;              // [64, 64, 16]

    float* acc = (float*)d_ws;                         // [B,O,16]  256 KB
    float* sp  = acc + (size_t)B_ * O_ * DOUT;         // [NCHUNK,B,O,16] 8 MB

    const int accn = B_ * O_ * DOUT;
    caps_zero<<<dim3((accn + 255) / 256), dim3(256), 0, stream>>>(acc, accn);

    for (int t = 0; t < 3; ++t) {
        caps_route_pass<<<dim3(NCHUNK, B_ / BB), dim3(NTHREADS), 0, stream>>>(
            x, w, acc, sp);
        caps_reduce_squash<<<dim3((B_ * O_) / 256), dim3(256), 0, stream>>>(
            sp, acc, outp, (t == 2) ? 1 : 0);
    }
}